// PatchFlow_75531294867782
// MI455X (gfx1250) — compile-verified
//
#include <hip/hip_runtime.h>
#include <math.h>

// ---- problem constants (match reference) ----
#define C_      32
#define H_      128
#define W_      128
#define B_      128
#define COND_   32
#define K_      7
#define S_      4
#define PAD_    2
#define HP_     32
#define WP_     32
#define NP_     (HP_ * WP_)          // 1024
#define DIN_    48                   // SPLIT1 + COND
#define HID_    96
#define DOUT_   32
#define NROWS_  (NP_ * B_)           // 131072
#define WPB     4                    // waves per block in flow kernel

typedef __attribute__((ext_vector_type(2))) float v2f;
typedef __attribute__((ext_vector_type(8))) float v8f;

__device__ __forceinline__ v8f wmma_f32(v2f a, v2f b, v8f c) {
  // D = A(16x4,f32) * B(4x16,f32) + C(16x16,f32)
  return __builtin_amdgcn_wmma_f32_16x16x4_f32(
      /*neg_a=*/false, a, /*neg_b=*/false, b,
      /*c_mod=*/(short)0, c, /*reuse_a=*/false, /*reuse_b=*/false);
}

// ---------------------------------------------------------------------------
// Kernel 1: local sqrt-mean-square pooling.
// One workgroup per (b, c) plane; the whole 128x128 plane (squared) is staged
// in LDS (64 KB of the 320 KB WGP pool), then each 32x32 output is a 7x7 sum.
// Output is written directly in the flow's row layout: row=(hp*WP+wp)*B + b.
// ---------------------------------------------------------------------------
__global__ void __launch_bounds__(256) pool_kernel(const float* __restrict__ x,
                                                   float* __restrict__ z0) {
  __shared__ float sq[H_ * W_];                 // 64 KB
  const int plane = blockIdx.x;                 // b * C + c
  const int bimg = plane / C_;
  const int ch   = plane % C_;
  const float* src = x + (size_t)plane * (H_ * W_);

  for (int i = threadIdx.x; i < H_ * W_; i += blockDim.x) {
    float v = src[i];
    sq[i] = v * v;
  }
  __syncthreads();

  for (int o = threadIdx.x; o < HP_ * WP_; o += blockDim.x) {
    const int hp = o / WP_, wp = o % WP_;
    const int h0 = hp * S_ - PAD_, w0 = wp * S_ - PAD_;
    float s = 0.f;
    #pragma unroll
    for (int kh = 0; kh < K_; ++kh) {
      const int h = h0 + kh;
      if (h < 0 || h >= H_) continue;
      #pragma unroll
      for (int kw = 0; kw < K_; ++kw) {
        const int w = w0 + kw;
        if (w < 0 || w >= W_) continue;
        s += sq[h * W_ + w];
      }
    }
    const float val = sqrtf(s * (1.0f / (K_ * K_)));
    z0[((size_t)(hp * WP_ + wp) * B_ + bimg) * C_ + ch] = val;
  }
}

// ---------------------------------------------------------------------------
// Kernel 2: fused 2-block conditional flow. One wave per 16-row tile.
// All GEMMs use V_WMMA_F32_16X16X4_F32; C/D->A fragment relayout goes through
// per-wave LDS staging (same-wave DS ops are in-order on CDNA5, no barriers).
// ---------------------------------------------------------------------------
__global__ void __launch_bounds__(32 * WPB) flow_kernel(
    const float* __restrict__ z0, const float* __restrict__ pos_enc,
    const float* __restrict__ w_perm, const float* __restrict__ g_scale,
    const float* __restrict__ g_offset, const float* __restrict__ W1,
    const float* __restrict__ b1, const float* __restrict__ W2,
    const float* __restrict__ b2, float* __restrict__ out_z,
    float* __restrict__ out_ldj) {
  __shared__ float st[WPB][16][C_];     // current z tile (16x32)       8 KB
  __shared__ float ht[WPB][16][HID_];   // h / y staging   (16x96)     24 KB

  const int lane = threadIdx.x & 31;
  const int wv   = threadIdx.x >> 5;
  const int tile = blockIdx.x * WPB + wv;
  const int row0 = tile * 16;
  const int p    = row0 / B_;           // patch index, constant per tile
  const int hi   = lane >> 4;           // lane group (0: lanes 0-15, 1: 16-31)
  const int ln   = lane & 15;

  // stage 0: load z tile into LDS (row-major 16x32)
  for (int i = lane; i < 16 * C_; i += 32) {
    const int m = i >> 5, cc = i & (C_ - 1);
    st[wv][m][cc] = z0[(size_t)(row0 + m) * C_ + cc];
  }

  // conditioner A-fragments (GEMM1 k-steps 4..11 cover cond cols 0..31);
  // identical for every row of the tile, so load once.
  v2f condA[8];
  #pragma unroll
  for (int ks = 0; ks < 8; ++ks) {
    const int kc = (ks + 4) * 4 + hi * 2 - 16;   // 0..31
    condA[ks].x = pos_enc[p * COND_ + kc];
    condA[ks].y = pos_enc[p * COND_ + kc + 1];
  }

  float ldj_acc[8];
  #pragma unroll
  for (int i = 0; i < 8; ++i) ldj_acc[i] = 0.f;

  for (int k = 0; k < 2; ++k) {
    const float* W1k = W1 + (size_t)k * HID_ * DIN_;
    const float* W2k = W2 + (size_t)k * DOUT_ * HID_;
    const float* wpk = w_perm + (size_t)k * C_ * C_;

    // ---- GEMM1: h = relu([x1 | cond] @ W1^T + b1)   (16x48 @ 48x96)
    v8f hfrag[6];
    #pragma unroll
    for (int nt = 0; nt < 6; ++nt) {
      const int n = nt * 16 + ln;
      const float bias = b1[k * HID_ + n];
      v8f acc;
      #pragma unroll
      for (int i = 0; i < 8; ++i) acc[i] = bias;
      #pragma unroll
      for (int ks = 0; ks < 12; ++ks) {
        const int k0 = ks * 4 + hi * 2;
        v2f a, b;
        if (ks < 4) { a.x = st[wv][ln][k0]; a.y = st[wv][ln][k0 + 1]; }
        else        { a = condA[ks - 4]; }
        b.x = W1k[n * DIN_ + k0];
        b.y = W1k[n * DIN_ + k0 + 1];
        acc = wmma_f32(a, b, acc);
      }
      #pragma unroll
      for (int i = 0; i < 8; ++i) acc[i] = fmaxf(acc[i], 0.f);
      hfrag[nt] = acc;
    }
    // restage h (C/D layout) -> LDS row-major for use as A
    #pragma unroll
    for (int nt = 0; nt < 6; ++nt)
      #pragma unroll
      for (int i = 0; i < 8; ++i)
        ht[wv][i + 8 * hi][nt * 16 + ln] = hfrag[nt][i];

    // ---- GEMM2: a = (h @ W2^T + b2) * 0.1           (16x96 @ 96x32)
    v8f afrag[2];
    #pragma unroll
    for (int nt = 0; nt < 2; ++nt) {
      const int n = nt * 16 + ln;
      const float bias = b2[k * DOUT_ + n];
      v8f acc;
      #pragma unroll
      for (int i = 0; i < 8; ++i) acc[i] = bias;
      #pragma unroll
      for (int ks = 0; ks < 24; ++ks) {
        const int k0 = ks * 4 + hi * 2;
        v2f a, b;
        a.x = ht[wv][ln][k0];       a.y = ht[wv][ln][k0 + 1];
        b.x = W2k[n * HID_ + k0];   b.y = W2k[n * HID_ + k0 + 1];
        acc = wmma_f32(a, b, acc);
      }
      #pragma unroll
      for (int i = 0; i < 8; ++i) acc[i] *= 0.1f;
      afrag[nt] = acc;
    }

    // ---- elementwise coupling + jacobian + scale/offset (y staged to ht)
    const float sc1 = 0.2f * log1pf(expf(0.5f * g_scale[k * C_ + ln]));
    const float of1 = g_offset[k * C_ + ln];
    const float sc2 = 0.2f * log1pf(expf(0.5f * g_scale[k * C_ + 16 + ln]));
    const float of2 = g_offset[k * C_ + 16 + ln];
    float slog = logf(sc1) + logf(sc2);
    #pragma unroll
    for (int off = 1; off < 16; off <<= 1) slog += __shfl_xor(slog, off, 32);

    #pragma unroll
    for (int i = 0; i < 8; ++i) {
      const int m = i + 8 * hi;
      const float s = 2.0f * 0.636f * atanf(afrag[0][i]);   // CLAMP*0.636*atan
      const float t = afrag[1][i];
      const float x2n = st[wv][m][16 + ln] * expf(s) + t;
      float srow = s;
      #pragma unroll
      for (int off = 1; off < 16; off <<= 1) srow += __shfl_xor(srow, off, 32);
      ldj_acc[i] += srow + slog;
      ht[wv][m][ln]      = st[wv][m][ln] * sc1 + of1;       // y: x1 half
      ht[wv][m][16 + ln] = x2n * sc2 + of2;                 // y: x2 half
    }

    // ---- permutation GEMM: z_new = y @ w_perm^T      (16x32 @ 32x32)
    v8f zfrag[2];
    #pragma unroll
    for (int nt = 0; nt < 2; ++nt) {
      const int n = nt * 16 + ln;
      v8f acc;
      #pragma unroll
      for (int i = 0; i < 8; ++i) acc[i] = 0.f;
      #pragma unroll
      for (int ks = 0; ks < 8; ++ks) {
        const int k0 = ks * 4 + hi * 2;
        v2f a, b;
        a.x = ht[wv][ln][k0];     a.y = ht[wv][ln][k0 + 1];
        b.x = wpk[n * C_ + k0];   b.y = wpk[n * C_ + k0 + 1];
        acc = wmma_f32(a, b, acc);
      }
      zfrag[nt] = acc;
    }
    // write back to st for next block (and to global on last block)
    #pragma unroll
    for (int nt = 0; nt < 2; ++nt)
      #pragma unroll
      for (int i = 0; i < 8; ++i) {
        const int m = i + 8 * hi;
        st[wv][m][nt * 16 + ln] = zfrag[nt][i];
        if (k == 1)
          out_z[(size_t)(row0 + m) * C_ + nt * 16 + ln] = zfrag[nt][i];
      }
  }

  if (ln == 0) {
    #pragma unroll
    for (int i = 0; i < 8; ++i)
      out_ldj[row0 + i + 8 * hi] = ldj_acc[i];
  }
}

// ---------------------------------------------------------------------------
extern "C" void kernel_launch(void* const* d_in, const int* in_sizes, int n_in,
                              void* d_out, int out_size, void* d_ws,
                              size_t ws_size, hipStream_t stream) {
  (void)in_sizes; (void)n_in; (void)out_size; (void)ws_size;
  const float* x        = (const float*)d_in[0];
  const float* pos_enc  = (const float*)d_in[1];
  const float* w_perm   = (const float*)d_in[2];
  const float* g_scale  = (const float*)d_in[3];
  const float* g_offset = (const float*)d_in[4];
  const float* W1       = (const float*)d_in[5];
  const float* b1       = (const float*)d_in[6];
  const float* W2       = (const float*)d_in[7];
  const float* b2       = (const float*)d_in[8];

  float* out_z   = (float*)d_out;                    // (NP, B, C)
  float* out_ldj = out_z + (size_t)NROWS_ * C_;      // (NP, B)
  float* z0      = (float*)d_ws;                     // pooled rows, 16 MB

  hipLaunchKernelGGL(pool_kernel, dim3(B_ * C_), dim3(256), 0, stream, x, z0);
  hipLaunchKernelGGL(flow_kernel, dim3(NROWS_ / 16 / WPB), dim3(32 * WPB), 0,
                     stream, z0, pos_enc, w_perm, g_scale, g_offset, W1, b1,
                     W2, b2, out_z, out_ldj);
}